// QuConv_6794638263059
// MI455X (gfx1250) — compile-verified
//
#include <hip/hip_runtime.h>
#include <hip/hip_bf16.h>
#include <math.h>

typedef float v2f __attribute__((ext_vector_type(2)));
typedef float v8f __attribute__((ext_vector_type(8)));

__device__ inline v8f wmma_f32(v2f a, v2f b, v8f c) {
  // V_WMMA_F32_16X16X4_F32 : D = A(16x4) * B(4x16) + C(16x16)
  return __builtin_amdgcn_wmma_f32_16x16x4_f32(false, a, false, b, (short)0, c,
                                               false, false);
}

// Async global->LDS copy helpers (CDNA5 GLOBAL_LOAD_ASYNC_TO_LDS_*, ASYNCcnt).
__device__ inline void async_copy_b128(unsigned lds_off, const void* gptr) {
  asm volatile("global_load_async_to_lds_b128 %0, %1, off"
               :: "v"(lds_off), "v"((unsigned long long)(uintptr_t)gptr)
               : "memory");
}
__device__ inline void async_copy_b64(unsigned lds_off, const void* gptr) {
  asm volatile("global_load_async_to_lds_b64 %0, %1, off"
               :: "v"(lds_off), "v"((unsigned long long)(uintptr_t)gptr)
               : "memory");
}
__device__ inline void wait_async0() {
  asm volatile("s_wait_asynccnt 0x0" ::: "memory");
}

// ---------------------------------------------------------------------------
// Gate construction: Up (4x4 complex) from 5 weights, then A2 = Up (x) Up.
// gate[0..255] = Re(A2) row-major, gate[256..511] = Im(A2).
// ---------------------------------------------------------------------------
struct Cpx { float re, im; };
__device__ inline Cpx cmul(Cpx a, Cpx b) {
  return {a.re * b.re - a.im * b.im, a.re * b.im + a.im * b.re};
}
__device__ inline Cpx cadd(Cpx a, Cpx b) { return {a.re + b.re, a.im + b.im}; }

__device__ inline void rx_gate(float t, Cpx m[2][2]) {
  float c = cosf(t * 0.5f), s = sinf(t * 0.5f);
  m[0][0] = {c, 0.f}; m[0][1] = {0.f, -s};
  m[1][0] = {0.f, -s}; m[1][1] = {c, 0.f};
}

__global__ void build_gate(const float* __restrict__ w, float* __restrict__ gate) {
  if (threadIdx.x != 0 || blockIdx.x != 0) return;
  const float WM = 0.63245553203367586f;  // sqrt(2)/sqrt(5)
  float th[5];
  for (int i = 0; i < 5; ++i) th[i] = w[i] * WM;

  Cpx r0[2][2], r1[2][2], rx01[2][2];
  rx_gate(th[0], r0);
  rx_gate(th[1], r1);
  for (int i = 0; i < 2; ++i)
    for (int j = 0; j < 2; ++j) {
      Cpx acc = {0.f, 0.f};
      for (int k = 0; k < 2; ++k) acc = cadd(acc, cmul(r1[i][k], r0[k][j]));
      rx01[i][j] = acc;
    }

  float c3 = cosf(th[3] * 0.5f), s3 = sinf(th[3] * 0.5f);
  float c4 = cosf(th[4] * 0.5f), s4 = sinf(th[4] * 0.5f);
  Cpx z3[2] = {{c3, -s3}, {c3, s3}};
  Cpx z4[2] = {{c4, -s4}, {c4, s4}};
  Cpx rz34[4];  // diagonal of kron(rz(th3), rz(th4))
  for (int i = 0; i < 2; ++i)
    for (int k = 0; k < 2; ++k) rz34[2 * i + k] = cmul(z3[i], z4[k]);

  float c2 = cosf(th[2] * 0.5f), s2 = sinf(th[2] * 0.5f);
  Cpx ryy[4][4];
  for (int i = 0; i < 4; ++i)
    for (int j = 0; j < 4; ++j) ryy[i][j] = {0.f, 0.f};
  for (int i = 0; i < 4; ++i) ryy[i][i] = {c2, 0.f};
  ryy[0][3] = {0.f, s2};  ryy[1][2] = {0.f, -s2};
  ryy[2][1] = {0.f, -s2}; ryy[3][0] = {0.f, s2};

  Cpx kx[4][4];  // kron(rx01, I2)
  for (int i = 0; i < 4; ++i)
    for (int j = 0; j < 4; ++j)
      kx[i][j] = ((i & 1) == (j & 1)) ? rx01[i >> 1][j >> 1] : Cpx{0.f, 0.f};

  Cpx Up[4][4];  // rz34 @ ryy @ kx
  for (int i = 0; i < 4; ++i)
    for (int j = 0; j < 4; ++j) {
      Cpx acc = {0.f, 0.f};
      for (int k = 0; k < 4; ++k) acc = cadd(acc, cmul(ryy[i][k], kx[k][j]));
      Up[i][j] = cmul(rz34[i], acc);
    }

  for (int i = 0; i < 16; ++i)
    for (int j = 0; j < 16; ++j) {
      Cpx v = cmul(Up[i >> 2][j >> 2], Up[i & 3][j & 3]);
      gate[i * 16 + j] = v.re;
      gate[256 + i * 16 + j] = v.im;
    }
}

// ---------------------------------------------------------------------------
// Left sweeps: out[base+m*s, col] = sum_k A2[m,k] * in[base+k*s, col]
// Block: 16-row group x 128 columns; wave w owns 16 columns (one WMMA tile).
// ---------------------------------------------------------------------------
template <bool REAL_IN>
__global__ __launch_bounds__(256) void apply_rows(const void* __restrict__ inp,
                                                  float2* __restrict__ out,
                                                  const float* __restrict__ gate,
                                                  int strideHi, int strideLo, int s) {
  __shared__ float gre[16 * 17], gim[16 * 17], gimn[16 * 17];
  __shared__ float2 tile[16][130];

  const int tid = threadIdx.x;
  {
    int m = tid >> 4, k = tid & 15;
    float r = gate[tid], i = gate[256 + tid];
    gre[m * 17 + k] = r;
    gim[m * 17 + k] = i;
    gimn[m * 17 + k] = -i;
  }
  const int col0 = blockIdx.x * 128;
  const int g = blockIdx.y;
  const size_t boff = (size_t)blockIdx.z * 4096u * 4096u;
  const int rowbase = (g >> 4) * strideHi + (g & 15) * strideLo;

  if (REAL_IN) {
    // First pass: expand real f32 input to interleaved complex in LDS.
    const float* X = (const float*)inp;
    for (int e = 0; e < 8; ++e) {
      int idx = tid + e * 256;
      int k = idx >> 7, n = idx & 127;
      size_t row = (size_t)rowbase + (size_t)k * s;
      tile[k][n] = make_float2(X[boff + row * 4096u + col0 + n], 0.f);
    }
  } else {
    // Complex passes: async DMA straight into LDS, 16B chunks (2 complex).
    const float2* X = (const float2*)inp;
    const unsigned ldsbase = (unsigned)(uintptr_t)(&tile[0][0]);
    for (int e = 0; e < 4; ++e) {
      int c = tid + e * 256;        // 16-byte chunk id, 0..1023
      int k = c >> 6;               // 64 chunks per 128-col row
      int n2 = (c & 63) * 2;        // first complex column of chunk
      size_t row = (size_t)rowbase + (size_t)k * s;
      async_copy_b128(ldsbase + (unsigned)((k * 130 + n2) * sizeof(float2)),
                      X + boff + row * 4096u + col0 + n2);
    }
    wait_async0();
  }
  __syncthreads();

  const int lane = tid & 31;
  const int n0 = (tid >> 5) * 16;        // wave's column slice
  const int mm = lane & 15;              // A row / B col / D col index
  const int khalf = (lane >> 4) << 1;    // 0 or 2

  v8f cre = {}, cim = {};
#pragma unroll
  for (int q = 0; q < 4; ++q) {
    int k0 = q * 4 + khalf;
    v2f ar, ai, an, br = {}, bi = {};
    ar.x = gre[mm * 17 + k0];  ar.y = gre[mm * 17 + k0 + 1];
    ai.x = gim[mm * 17 + k0];  ai.y = gim[mm * 17 + k0 + 1];
    an.x = gimn[mm * 17 + k0]; an.y = gimn[mm * 17 + k0 + 1];
    float2 t0 = tile[k0][n0 + mm];
    float2 t1 = tile[k0 + 1][n0 + mm];
    br.x = t0.x; br.y = t1.x;
    cre = wmma_f32(ar, br, cre);   // Re += Ar*Br
    cim = wmma_f32(ai, br, cim);   // Im += Ai*Br
    if (!REAL_IN) {
      bi.x = t0.y; bi.y = t1.y;
      cre = wmma_f32(an, bi, cre); // Re += (-Ai)*Bi
      cim = wmma_f32(ar, bi, cim); // Im += Ar*Bi
    }
  }

  const int mhi = (lane >> 4) << 3;
#pragma unroll
  for (int i = 0; i < 8; ++i) {
    int m = i + mhi;
    size_t row = (size_t)rowbase + (size_t)m * s;
    out[boff + row * 4096u + col0 + n0 + mm] = make_float2(cre[i], cim[i]);
  }
}

// ---------------------------------------------------------------------------
// Right sweeps: out[r, base+j*s] = sum_k in[r, base+k*s] * conj(A2[j,k])
// Block: 128 rows x one 16-column group; wave w owns rows [16w, 16w+16).
// ---------------------------------------------------------------------------
__global__ __launch_bounds__(256) void apply_cols(float2* __restrict__ io,
                                                  const float* __restrict__ gate,
                                                  int strideHi, int strideLo, int s) {
  __shared__ float gret[16 * 17], gimt[16 * 17], gimnt[16 * 17];
  __shared__ float2 tile[128][17];

  const int tid = threadIdx.x;
  {
    int m = tid >> 4, k = tid & 15;        // A2[m,k]
    float r = gate[tid], i = gate[256 + tid];
    gret[k * 17 + m] = r;                  // (A2^T)re  -> Hr
    gimt[k * 17 + m] = i;                  // (A2^T)im  -> -Hi
    gimnt[k * 17 + m] = -i;                // -(A2^T)im ->  Hi
  }
  const int row0 = blockIdx.x * 128;
  const int g = blockIdx.y;
  const size_t boff = (size_t)blockIdx.z * 4096u * 4096u;
  const int colbase = (g >> 4) * strideHi + (g & 15) * strideLo;

  const unsigned ldsbase = (unsigned)(uintptr_t)(&tile[0][0]);
  if (s == 1) {
    // Contiguous columns: 16B async chunks (2 complex per chunk).
    for (int e = 0; e < 4; ++e) {
      int c = tid + e * 256;        // chunk id, 0..1023
      int r = c >> 3;               // 8 chunks per 16-col row
      int k2 = (c & 7) * 2;
      async_copy_b128(ldsbase + (unsigned)((r * 17 + k2) * sizeof(float2)),
                      io + boff + (size_t)(row0 + r) * 4096u + colbase + k2);
    }
  } else {
    // Strided columns: 8B async chunks (one complex each).
    for (int e = 0; e < 8; ++e) {
      int idx = tid + e * 256;
      int r = idx >> 4, k = idx & 15;
      async_copy_b64(ldsbase + (unsigned)((r * 17 + k) * sizeof(float2)),
                     io + boff + (size_t)(row0 + r) * 4096u + colbase + k * s);
    }
  }
  wait_async0();
  __syncthreads();

  const int lane = tid & 31;
  const int wv = tid >> 5;
  const int mm = lane & 15;
  const int khalf = (lane >> 4) << 1;
  const int rloc = wv * 16 + mm;

  v8f cre = {}, cim = {};
#pragma unroll
  for (int q = 0; q < 4; ++q) {
    int k0 = q * 4 + khalf;
    v2f xr, xi, hr, hi, hin;
    float2 t0 = tile[rloc][k0];
    float2 t1 = tile[rloc][k0 + 1];
    xr.x = t0.x; xr.y = t1.x;
    xi.x = t0.y; xi.y = t1.y;
    hr.x  = gret[k0 * 17 + mm];  hr.y  = gret[(k0 + 1) * 17 + mm];
    hi.x  = gimnt[k0 * 17 + mm]; hi.y  = gimnt[(k0 + 1) * 17 + mm];
    hin.x = gimt[k0 * 17 + mm];  hin.y = gimt[(k0 + 1) * 17 + mm];
    cre = wmma_f32(xr, hr, cre);   // Re += Xr*Hr
    cre = wmma_f32(xi, hin, cre);  // Re += Xi*(-Hi)
    cim = wmma_f32(xr, hi, cim);   // Im += Xr*Hi
    cim = wmma_f32(xi, hr, cim);   // Im += Xi*Hr
  }

  const int mhi = (lane >> 4) << 3;
#pragma unroll
  for (int i = 0; i < 8; ++i) {
    int r = wv * 16 + i + mhi;
    io[boff + (size_t)(row0 + r) * 4096u + colbase + mm * s] =
        make_float2(cre[i], cim[i]);
  }
}

// ---------------------------------------------------------------------------
extern "C" void kernel_launch(void* const* d_in, const int* in_sizes, int n_in,
                              void* d_out, int out_size, void* d_ws, size_t ws_size,
                              hipStream_t stream) {
  (void)in_sizes; (void)n_in; (void)out_size; (void)ws_size;
  const float* x = (const float*)d_in[0];
  const float* w = (const float*)d_in[1];
  float* gate = (float*)d_ws;          // 512 floats of scratch
  float2* out = (float2*)d_out;        // complex64, interleaved re/im

  build_gate<<<1, 64, 0, stream>>>(w, gate);

  dim3 blk(256);
  dim3 grd(32, 256, 4);  // (tile, group, batch)
  // Left: U @ X  — contract row digits t (s=1), q (s=16), p (s=256)
  apply_rows<true><<<grd, blk, 0, stream>>>(x, out, gate, 256, 16, 1);
  apply_rows<false><<<grd, blk, 0, stream>>>(out, out, gate, 256, 1, 16);
  apply_rows<false><<<grd, blk, 0, stream>>>(out, out, gate, 16, 1, 256);
  // Right: (U X) @ U^H — contract col digits
  apply_cols<<<grd, blk, 0, stream>>>(out, gate, 256, 16, 1);
  apply_cols<<<grd, blk, 0, stream>>>(out, gate, 256, 1, 16);
  apply_cols<<<grd, blk, 0, stream>>>(out, gate, 16, 1, 256);
}